// GAT_17068200034947
// MI455X (gfx1250) — compile-verified
//
#include <hip/hip_runtime.h>

#define N_NODES 250000
#define N_EDGES 3000000
#define KEEP    187215
#define NEG_SLOPE 0.2f

typedef float v2f __attribute__((ext_vector_type(2)));
typedef float v8f __attribute__((ext_vector_type(8)));

__device__ __forceinline__ float warpSum32(float v) {
#pragma unroll
  for (int off = 16; off > 0; off >>= 1) v += __shfl_xor(v, off, 32);
  return v;
}

__device__ __forceinline__ void atomicMaxF(float* addr, float v) {
  if (v >= 0.0f) atomicMax((int*)addr, __float_as_int(v));
  else           atomicMin((unsigned int*)addr, __float_as_uint(v));
}

__device__ __forceinline__ unsigned f2key(float f) {
  unsigned u = __float_as_uint(f);
  return (u & 0x80000000u) ? ~u : (u | 0x80000000u);  // monotone: bigger float -> bigger key
}

__device__ __forceinline__ float lrelu(float v) { return v > 0.0f ? v : NEG_SLOPE * v; }

// ---------------- K1: h = x @ W1 via V_WMMA_F32_16X16X4_F32 ----------------
// One wave per 16x16 output tile; K=64 -> 16 chained WMMAs. 4 waves/block cover N dim.
__global__ void k_gemm_h(const float* __restrict__ x, const float* __restrict__ W1,
                         float* __restrict__ h) {
  const int lane  = threadIdx.x & 31;
  const int wave  = threadIdx.x >> 5;          // 0..3 -> n-tile
  const int m0    = blockIdx.x * 16;
  const int n0    = wave * 16;
  const int r16   = lane & 15;
  const int khalf = (lane >> 4) * 2;           // lanes 16-31 hold K+2/K+3
  v8f acc = {};
  const float* xrow = x + (size_t)(m0 + r16) * 64 + khalf;
#pragma unroll
  for (int kk = 0; kk < 16; ++kk) {
    const int k0 = kk * 4;
    v2f a = *(const v2f*)(xrow + k0);          // A 16x4: lane=row, vgpr pair = K
    v2f b;                                     // B 4x16: lane=col, vgpr pair = K
    b.x = W1[(size_t)(k0 + khalf) * 64 + n0 + r16];
    b.y = W1[(size_t)(k0 + khalf + 1) * 64 + n0 + r16];
    acc = __builtin_amdgcn_wmma_f32_16x16x4_f32(false, a, false, b, (short)0, acc,
                                                false, false);
  }
  // C/D layout: VGPR r -> M = r (lanes 0-15) / 8+r (lanes 16-31), N = lane&15
  float* hp = h + (size_t)(m0 + (lane >> 4) * 8) * 64 + n0 + r16;
#pragma unroll
  for (int r = 0; r < 8; ++r) hp[(size_t)r * 64] = acc[r];
}

// ---------------- K2: a_src/a_dst = (h * att).sum(-1), wave per node ----------------
__global__ void k_attn_coef(const float* __restrict__ h, const float* __restrict__ att_src,
                            const float* __restrict__ att_dst,
                            float* __restrict__ asrc, float* __restrict__ adst) {
  const int i    = (blockIdx.x * blockDim.x + threadIdx.x) >> 5;
  const int lane = threadIdx.x & 31;
  const float h0 = h[(size_t)i * 64 + lane];
  const float h1 = h[(size_t)i * 64 + 32 + lane];
  const float s0 = warpSum32(h0 * att_src[lane]);
  const float s1 = warpSum32(h1 * att_src[32 + lane]);
  const float d0 = warpSum32(h0 * att_dst[lane]);
  const float d1 = warpSum32(h1 * att_dst[32 + lane]);
  if (lane == 0) { asrc[2*i] = s0; asrc[2*i+1] = s1; adst[2*i] = d0; adst[2*i+1] = d1; }
}

// ---------------- K3: m init with self-loop logit, zero neigh ----------------
__global__ void k_init_m(const float* __restrict__ asrc, const float* __restrict__ adst,
                         float* __restrict__ m, float* __restrict__ neigh) {
  const int i = blockIdx.x * blockDim.x + threadIdx.x;
  if (i >= N_NODES) return;
  m[2*i]   = lrelu(asrc[2*i]   + adst[2*i]);
  m[2*i+1] = lrelu(asrc[2*i+1] + adst[2*i+1]);
  neigh[i] = 0.0f;
}

// ---------------- K4: segment_max over edges (atomic float max) ----------------
__global__ void k_edge_max(const int* __restrict__ ei, const float* __restrict__ asrc,
                           const float* __restrict__ adst, float* __restrict__ m) {
  const int e = blockIdx.x * blockDim.x + threadIdx.x;
  if (e >= N_EDGES) return;
  const int s = ei[e], d = ei[N_EDGES + e];
  atomicMaxF(&m[2*d],   lrelu(asrc[2*s]   + adst[2*d]));
  atomicMaxF(&m[2*d+1], lrelu(asrc[2*s+1] + adst[2*d+1]));
}

// ---------------- K5: self-loop term seeds denom and out numerator ----------------
__global__ void k_self_init(const float* __restrict__ h, const float* __restrict__ asrc,
                            const float* __restrict__ adst, const float* __restrict__ m,
                            float* __restrict__ out, float* __restrict__ denom) {
  const int i    = (blockIdx.x * blockDim.x + threadIdx.x) >> 5;
  const int lane = threadIdx.x & 31;
  const float w0 = __expf(lrelu(asrc[2*i]   + adst[2*i])   - m[2*i]);
  const float w1 = __expf(lrelu(asrc[2*i+1] + adst[2*i+1]) - m[2*i+1]);
  out[(size_t)i*64 + lane]      = w0 * h[(size_t)i*64 + lane];
  out[(size_t)i*64 + 32 + lane] = w1 * h[(size_t)i*64 + 32 + lane];
  if (lane == 0) { denom[2*i] = w0; denom[2*i+1] = w1; }
}

// ---------------- K6: edge scatter (numerator + denom), wave per edge ----------------
__global__ void k_edge_scatter(const int* __restrict__ ei, const float* __restrict__ asrc,
                               const float* __restrict__ adst, const float* __restrict__ m,
                               const float* __restrict__ h, float* __restrict__ out,
                               float* __restrict__ denom) {
  const int e    = (blockIdx.x * blockDim.x + threadIdx.x) >> 5;
  const int lane = threadIdx.x & 31;
  const int s = ei[e], d = ei[N_EDGES + e];
  const float w0 = __expf(lrelu(asrc[2*s]   + adst[2*d])   - m[2*d]);
  const float w1 = __expf(lrelu(asrc[2*s+1] + adst[2*d+1]) - m[2*d+1]);
  const float* hs = h + (size_t)s * 64;
  float* od = out + (size_t)d * 64;
  unsafeAtomicAdd(&od[lane],      w0 * hs[lane]);
  unsafeAtomicAdd(&od[32 + lane], w1 * hs[32 + lane]);
  if (lane == 0) unsafeAtomicAdd(&denom[2*d],   w0);
  if (lane == 1) unsafeAtomicAdd(&denom[2*d+1], w1);
}

// ---------------- K7: out = relu(num/denom + b1); fold Wr, Wn, Wl dots ----------------
__global__ void k_finalize(float* __restrict__ out, const float* __restrict__ denom,
                           const float* __restrict__ b1, const float* __restrict__ Wr,
                           const float* __restrict__ Wn, const float* __restrict__ Wl,
                           float* __restrict__ rsum, float* __restrict__ tn,
                           float* __restrict__ u) {
  const int i    = (blockIdx.x * blockDim.x + threadIdx.x) >> 5;
  const int lane = threadIdx.x & 31;
  const int c0 = lane, c1 = lane + 32;
  const float d0 = denom[2*i], d1 = denom[2*i+1];
  float o0 = out[(size_t)i*64 + c0] / d0 + b1[c0]; o0 = o0 > 0.0f ? o0 : 0.0f;
  float o1 = out[(size_t)i*64 + c1] / d1 + b1[c1]; o1 = o1 > 0.0f ? o1 : 0.0f;
  out[(size_t)i*64 + c0] = o0;
  out[(size_t)i*64 + c1] = o1;
  const float pr  = warpSum32(o0 * Wr[c0]      + o1 * Wr[c1]);
  const float pn  = warpSum32(o0 * Wn[c0]      + o1 * Wn[c1]);
  const float pl0 = warpSum32(o0 * Wl[c0*2]    + o1 * Wl[c1*2]);
  const float pl1 = warpSum32(o0 * Wl[c0*2+1]  + o1 * Wl[c1*2+1]);
  if (lane == 0) { rsum[i] = pr; tn[i] = pn; u[2*i] = pl0; u[2*i+1] = pl1; }
}

// ---------------- K8: neigh[d] += (out@Wn)[s] over original edges (scalar!) --------
__global__ void k_neigh(const int* __restrict__ ei, const float* __restrict__ tn,
                        float* __restrict__ neigh) {
  const int e = blockIdx.x * blockDim.x + threadIdx.x;
  if (e >= N_EDGES) return;
  unsafeAtomicAdd(&neigh[ei[N_EDGES + e]], tn[ei[e]]);
}

// ---------------- K9: score = tanh(.), accumulate softmax denominator --------------
__global__ void k_score(const float* __restrict__ rsum, const float* __restrict__ neigh,
                        const float* __restrict__ bg, float* __restrict__ score,
                        float* __restrict__ Ssum) {
  const int i = blockIdx.x * blockDim.x + threadIdx.x;
  float ex = 0.0f;
  if (i < N_NODES) {
    const float sc = tanhf(rsum[i] + neigh[i] + bg[0]);
    score[i] = sc;
    ex = __expf(sc);
  }
  const float s = warpSum32(ex);
  if ((threadIdx.x & 31) == 0) unsafeAtomicAdd(Ssum, s);
}

// ---------------- Radix select for top-KEEP threshold (3 rounds) -------------------
__global__ void k_sel_hist(const float* __restrict__ score, const unsigned* __restrict__ state,
                           unsigned* __restrict__ hist, int round) {
  __shared__ unsigned lh[2048];
  for (int b = threadIdx.x; b < 2048; b += blockDim.x) lh[b] = 0;
  __syncthreads();
  const int i = blockIdx.x * blockDim.x + threadIdx.x;
  if (i < N_NODES) {
    const unsigned key = f2key(score[i]);
    const unsigned prefix = state[0];
    if (round == 0)                          atomicAdd(&lh[key >> 21], 1u);
    else if (round == 1) { if ((key >> 21) == prefix) atomicAdd(&lh[(key >> 10) & 2047u], 1u); }
    else                 { if ((key >> 10) == prefix) atomicAdd(&lh[key & 1023u], 1u); }
  }
  __syncthreads();
  for (int b = threadIdx.x; b < 2048; b += blockDim.x)
    if (lh[b]) atomicAdd(&hist[b], lh[b]);
}

__global__ void k_sel_scan(unsigned* __restrict__ state, unsigned* __restrict__ hist, int round) {
  if (threadIdx.x == 0) {
    const unsigned K = state[1];
    const int nb = (round == 2) ? 1024 : 2048;
    unsigned cum = 0, digit = 0, Knew = K;
    for (int b = nb - 1; b >= 0; --b) {
      const unsigned c = hist[b];
      if (cum + c >= K) { digit = (unsigned)b; Knew = K - cum; break; }
      cum += c;
    }
    const unsigned shift = (round == 2) ? 10u : 11u;
    state[0] = (state[0] << shift) | digit;
    state[1] = Knew;
  }
  __syncthreads();
  for (int b = threadIdx.x; b < 2048; b += blockDim.x) hist[b] = 0;  // ready for next round
}

// ---------------- K10: masked weighted sum (top-k + mean-pool + Wl folded) ---------
__global__ void k_pool(const float* __restrict__ score, const float* __restrict__ u,
                       const unsigned* __restrict__ state, unsigned* __restrict__ ticket,
                       float* __restrict__ acc) {
  const int i = blockIdx.x * blockDim.x + threadIdx.x;
  float a0 = 0.0f, a1 = 0.0f;
  if (i < N_NODES) {
    const float sc = score[i];
    const unsigned key = f2key(sc);
    const unsigned T = state[0];
    bool inc = key > T;
    if (key == T) { const unsigned t = atomicAdd(ticket, 1u); inc = (t < state[1]); }
    if (inc) { a0 = sc * u[2*i]; a1 = sc * u[2*i+1]; }
  }
  a0 = warpSum32(a0);
  a1 = warpSum32(a1);
  if ((threadIdx.x & 31) == 0) { unsafeAtomicAdd(&acc[0], a0); unsafeAtomicAdd(&acc[1], a1); }
}

__global__ void k_logits(const float* __restrict__ acc, const float* __restrict__ bl,
                         float* __restrict__ dout) {
  if (threadIdx.x == 0) {
    dout[0] = acc[0] / (float)KEEP + bl[0];
    dout[1] = acc[1] / (float)KEEP + bl[1];
  }
}

__global__ void k_softmax_out(const float* __restrict__ score, const float* __restrict__ Ssum,
                              float* __restrict__ dout) {
  const int i = blockIdx.x * blockDim.x + threadIdx.x;
  if (i < N_NODES) dout[2 + i] = __expf(score[i]) / Ssum[0];
}

__global__ void k_init_ws(float* __restrict__ acc, float* __restrict__ Ssum,
                          unsigned* __restrict__ ticket, unsigned* __restrict__ hist,
                          unsigned* __restrict__ state) {
  const int t = threadIdx.x;
  if (t == 0) { acc[0] = 0.0f; acc[1] = 0.0f; Ssum[0] = 0.0f; ticket[0] = 0u;
                state[0] = 0u; state[1] = (unsigned)KEEP; }
  for (int b = t; b < 2048; b += blockDim.x) hist[b] = 0u;
}

extern "C" void kernel_launch(void* const* d_in, const int* in_sizes, int n_in,
                              void* d_out, int out_size, void* d_ws, size_t ws_size,
                              hipStream_t stream) {
  const float* x       = (const float*)d_in[0];
  const int*   ei      = (const int*)d_in[1];
  const float* W1      = (const float*)d_in[3];
  const float* att_src = (const float*)d_in[4];
  const float* att_dst = (const float*)d_in[5];
  const float* b1      = (const float*)d_in[6];
  const float* Wr      = (const float*)d_in[7];
  const float* Wn      = (const float*)d_in[8];
  const float* bg      = (const float*)d_in[9];
  const float* Wl      = (const float*)d_in[10];
  const float* bl      = (const float*)d_in[11];
  float* dout = (float*)d_out;

  float* ws_h     = (float*)d_ws;          // N*64
  float* ws_out   = ws_h     + 16000000;   // N*64
  float* ws_asrc  = ws_out   + 16000000;   // N*2
  float* ws_adst  = ws_asrc  + 500000;     // N*2
  float* ws_m     = ws_adst  + 500000;     // N*2
  float* ws_den   = ws_m     + 500000;     // N*2
  float* ws_rsum  = ws_den   + 500000;     // N
  float* ws_tn    = ws_rsum  + 250000;     // N
  float* ws_neigh = ws_tn    + 250000;     // N
  float* ws_score = ws_neigh + 250000;     // N
  float* ws_u     = ws_score + 250000;     // N*2
  float* ws_acc   = ws_u     + 500000;     // 2
  float* ws_ssum  = ws_acc   + 2;          // 1
  unsigned* ws_ticket = (unsigned*)(ws_ssum + 1);  // 1
  unsigned* ws_hist   = ws_ticket + 1;             // 2048
  unsigned* ws_state  = ws_hist + 2048;            // 2

  k_init_ws<<<1, 256, 0, stream>>>(ws_acc, ws_ssum, ws_ticket, ws_hist, ws_state);

  // GEMM: 15625 row-tiles of 16 (exact), 4 waves (n-tiles) per block
  k_gemm_h<<<15625, 128, 0, stream>>>(x, W1, ws_h);

  // attention coefficients: 8 waves/block, 31250*8 == N exactly
  k_attn_coef<<<31250, 256, 0, stream>>>(ws_h, att_src, att_dst, ws_asrc, ws_adst);

  k_init_m<<<(N_NODES + 255) / 256, 256, 0, stream>>>(ws_asrc, ws_adst, ws_m, ws_neigh);
  k_edge_max<<<(N_EDGES + 255) / 256, 256, 0, stream>>>(ei, ws_asrc, ws_adst, ws_m);
  k_self_init<<<31250, 256, 0, stream>>>(ws_h, ws_asrc, ws_adst, ws_m, ws_out, ws_den);
  // wave per edge: 375000*8 == E exactly
  k_edge_scatter<<<375000, 256, 0, stream>>>(ei, ws_asrc, ws_adst, ws_m, ws_h, ws_out, ws_den);
  k_finalize<<<31250, 256, 0, stream>>>(ws_out, ws_den, b1, Wr, Wn, Wl, ws_rsum, ws_tn, ws_u);
  k_neigh<<<(N_EDGES + 255) / 256, 256, 0, stream>>>(ei, ws_tn, ws_neigh);
  k_score<<<(N_NODES + 255) / 256, 256, 0, stream>>>(ws_rsum, ws_neigh, bg, ws_score, ws_ssum);

  for (int round = 0; round < 3; ++round) {
    k_sel_hist<<<(N_NODES + 255) / 256, 256, 0, stream>>>(ws_score, ws_state, ws_hist, round);
    k_sel_scan<<<1, 256, 0, stream>>>(ws_state, ws_hist, round);
  }

  k_pool<<<(N_NODES + 255) / 256, 256, 0, stream>>>(ws_score, ws_u, ws_state, ws_ticket, ws_acc);
  k_logits<<<1, 32, 0, stream>>>(ws_acc, bl, dout);
  k_softmax_out<<<(N_NODES + 255) / 256, 256, 0, stream>>>(ws_score, ws_ssum, dout);
}